// MPLayer_17566416240734
// MI455X (gfx1250) — compile-verified
//
#include <hip/hip_runtime.h>

// MI455X / gfx1250, wave32. D = A×B + C via V_WMMA_F32_16X16X4_F32.
typedef __attribute__((ext_vector_type(2))) float v2f;
typedef __attribute__((ext_vector_type(8))) float v8f;

#define NNODES 50000
#define KNBR   32
#define FDIM   128
#define EDIM   16
#define NODES_PER_WG 16
#define LSTRIDE 18                 // floats per l-column of S in LDS (pad 16->18)
#define SROW   (FDIM * LSTRIDE + 4)  // 2308 floats per node: conflict-free b64 reads

// Only check in the DEVICE pass; host pass doesn't see amdgcn builtins.
#if defined(__HIP_DEVICE_COMPILE__) && defined(__AMDGCN__)
#if !__has_builtin(__builtin_amdgcn_wmma_f32_16x16x4_f32)
#error "device toolchain lacks __builtin_amdgcn_wmma_f32_16x16x4_f32"
#endif
#endif

// ---- pre-pass: pair-interleaved transposed weights (1 MB into d_ws) ----
// kk = l*16 + n   (contraction index of stage 2)
// wtp[(kk>>1)*256 + m*2 + (kk&1)] = w[(l*128 + m)*16 + n]
// => a lane's B-fragment pair (rows kk, kk+1; same column m) is one b64,
//    and lanes 0-15 cover one contiguous 128B line.
__global__ void transpose_w_kernel(const float* __restrict__ w,
                                   float* __restrict__ wtp) {
  int t = blockIdx.x * blockDim.x + threadIdx.x;  // 0 .. 2048*128-1
  int parity = t & 1;
  int m = (t >> 1) & 127;
  int p = t >> 8;
  int kk = 2 * p + parity;
  int l = kk >> 4;
  int n = kk & 15;
  wtp[t] = w[(l * FDIM + m) * EDIM + n];
}

// ---- fused message-passing kernel: 1 WG (4 waves) = 16 nodes ----
__launch_bounds__(128, 1)
__global__ void mplayer_kernel(const float* __restrict__ nodes,
                               const int*   __restrict__ nlist,
                               const float* __restrict__ edges,
                               const float* __restrict__ w,
                               const float* __restrict__ wtp,
                               int use_wt,
                               float* __restrict__ out) {
  // S block: 16 nodes x (128 l-columns x 18) floats, kk = l*16 + n.
  // 147.7 KB of the WGP's 320 KB LDS -> 2 workgroups per WGP.
  __shared__ __align__(16) float sS[NODES_PER_WG * SROW];

  const int lane = threadIdx.x & 31;
  const int wave = threadIdx.x >> 5;
  const int col  = lane & 15;   // N-column within a 16-wide tile
  const int half = lane >> 4;   // lane half selects K rows / M rows per WMMA layout
  const int i0   = blockIdx.x * NODES_PER_WG;

  // ========== Stage 1: S_i[16x128] = edges_i^T (16x32) x H_i (32x128) ==========
  for (int q = 0; q < 4; ++q) {
    const int il = wave * 4 + q;       // node slot 0..15 in this WG
    const int i  = i0 + il;

    // lane j holds neighbor j's byte offset into `nodes` (50000*512 < 2^31)
    int off = nlist[i * KNBR + lane] * (FDIM * 4);

    const float* epi = edges + (size_t)i * (KNBR * EDIM);
    __builtin_prefetch(epi + KNBR * EDIM, 0, 1);   // next node's edge block

    // A = edges^T fragments (M rows = E dim, K = neighbor dim j).
    // 16x4 f32 A layout: lane<16 holds (K=j0, j0+1), lane>=16 (j0+2, j0+3).
    v2f A[8];
#pragma unroll
    for (int s = 0; s < 8; ++s) {
      int j0 = 4 * s + 2 * half;
      A[s].x = epi[j0 * EDIM + col];        // coalesced 64B across lanes 0-15
      A[s].y = epi[(j0 + 1) * EDIM + col];
    }
    // neighbor-row byte offsets each lane needs for its B-fragment rows
    int r0[8], r1[8];
#pragma unroll
    for (int s = 0; s < 8; ++s) {
      int j0 = 4 * s + 2 * half;
      r0[s] = __shfl(off, j0, 32);
      r1[s] = __shfl(off, j0 + 1, 32);
    }

    const char* nbase = (const char*)nodes;
#pragma unroll
    for (int nt = 0; nt < 8; ++nt) {      // 8 N-tiles cover F=128
      v8f acc = {};
      const int cb = (nt * 16 + col) * 4;
#pragma unroll
      for (int s = 0; s < 8; ++s) {       // K=32 in steps of 4
        v2f b;                            // B rows = gathered neighbor features
        b.x = *(const float*)(nbase + (size_t)(unsigned)r0[s] + cb);
        b.y = *(const float*)(nbase + (size_t)(unsigned)r1[s] + cb);
        acc = __builtin_amdgcn_wmma_f32_16x16x4_f32(
            false, A[s], false, b, (short)0, acc, false, false);
      }
      // C tile: lane owns column l = nt*16+col, rows n = 8*half .. 8*half+7
      // -> 8 contiguous floats in S (kk = l*16 + n): 4x ds_store_b64
      float* srow = &sS[il * SROW + (nt * 16 + col) * LSTRIDE + 8 * half];
#pragma unroll
      for (int r = 0; r < 8; r += 2) {
        v2f st; st.x = acc[r]; st.y = acc[r + 1];
        *(v2f*)(srow + r) = st;
      }
    }
  }
  __syncthreads();

  // ====== Stage 2: out[16x128] = S[16x2048] x W2[2048x128], K in steps of 4 ====
#pragma unroll
  for (int mt2 = 0; mt2 < 2; ++mt2) {
    const int mt    = wave * 2 + mt2;   // 4 waves x 2 = 8 m-tiles of 16
    const int mbase = mt * 16;
    v8f acc = {};
    // A rows = nodes; lane reads kk = c0+2h, c0+2h+1 (contiguous -> ds_load_b64)
    const float* sA   = &sS[(lane & 15) * SROW + 2 * half];
    const float* wppL = wtp + (size_t)half * 256 + (size_t)(mbase + col) * 2;
#pragma unroll 8
    for (int c0 = 0; c0 < EDIM * FDIM; c0 += 4) {
      v2f a = *(const v2f*)(sA + (c0 >> 4) * LSTRIDE + (c0 & 15));
      v2f b;
      if (use_wt) {                      // one coalesced b64 per lane
        b = *(const v2f*)(wppL + (size_t)(c0 >> 1) * 256);
      } else {                           // fallback: strided reads of original w
        int kk0 = c0 + 2 * half;
        int kk1 = kk0 + 1;
        b.x = w[((kk0 >> 4) * FDIM + mbase + col) * EDIM + (kk0 & 15)];
        b.y = w[((kk1 >> 4) * FDIM + mbase + col) * EDIM + (kk1 & 15)];
      }
      acc = __builtin_amdgcn_wmma_f32_16x16x4_f32(
          false, a, false, b, (short)0, acc, false, false);
    }
    const float scale = 1.0f / (float)KNBR;   // mean over neighbors
    float* op = out + (size_t)(i0 + 8 * half) * FDIM + mbase + col;
#pragma unroll
    for (int r = 0; r < 8; ++r)
      op[(size_t)r * FDIM] = acc[r] * scale;
  }
}

extern "C" void kernel_launch(void* const* d_in, const int* in_sizes, int n_in,
                              void* d_out, int out_size, void* d_ws, size_t ws_size,
                              hipStream_t stream) {
  const float* nodes = (const float*)d_in[0];
  const int*   nlist = (const int*)d_in[1];
  const float* edges = (const float*)d_in[2];
  const float* w     = (const float*)d_in[3];
  float* out = (float*)d_out;
  float* wtp = (float*)d_ws;

  const size_t wt_bytes = (size_t)EDIM * FDIM * FDIM * sizeof(float);  // 1 MB
  int use_wt = (ws_size >= wt_bytes) ? 1 : 0;
  if (use_wt) {
    transpose_w_kernel<<<(EDIM * FDIM * FDIM) / 256, 256, 0, stream>>>(w, wtp);
  }
  mplayer_kernel<<<NNODES / NODES_PER_WG, 128, 0, stream>>>(
      nodes, nlist, edges, w, wtp, use_wt, out);
}